// SGNN_64716567216291
// MI455X (gfx1250) — compile-verified
//
#include <hip/hip_runtime.h>
#include <math.h>

// ---------------------------------------------------------------------------
// SGNN forward for MI455X (gfx1250, wave32).
// All dense algebra runs through V_WMMA_F32_16X16X4_F32 (full fp32 fidelity;
// end-to-end time is bound by the L2-resident edge aggregation, so low
// precision WMMA would not help).
// ---------------------------------------------------------------------------

typedef __attribute__((ext_vector_type(2))) float v2f;
typedef __attribute__((ext_vector_type(8))) float v8f;

#define NODES 50000
#define FEAT  256
#define HID   128
#define CLS   40
#define NLAYER 4

// ----------------------------- helpers -------------------------------------

__global__ void zero_kernel(float* __restrict__ buf, int n) {
  int i = blockIdx.x * blockDim.x + threadIdx.x;
  if (i < n) buf[i] = 0.0f;
}

// agg[dst[e]] += v[src[e]]  for all edges, 128 lanes per edge (coalesced row).
// Atomics land in L2 (working set << 192MB L2).
__global__ void scatter_add_kernel(const float* __restrict__ v,
                                   const int* __restrict__ src,
                                   const int* __restrict__ dst,
                                   float* __restrict__ agg, int E) {
  long long t = (long long)blockIdx.x * blockDim.x + threadIdx.x;
  int e = (int)(t >> 7);        // HID == 128
  int h = (int)(t & 127);
  if (e < E) {
    int s = src[e];
    int d = dst[e];
    atomicAdd(&agg[(long long)d * HID + h], v[(long long)s * HID + h]);
  }
}

// symW[2*i]   = 0.5*(up_Ws[i] + up_Ws[i]^T)
// symW[2*i+1] = 0.5*(down_Ws[i] + down_Ws[i]^T)
__global__ void symmetrize_kernel(const float* __restrict__ upW,
                                  const float* __restrict__ dnW,
                                  float* __restrict__ symW) {
  int t = blockIdx.x * blockDim.x + threadIdx.x;
  const int HH = HID * HID;
  if (t < (NLAYER + 1) * HH) {
    int i  = t / HH;
    int rc = t % HH;
    int r = rc / HID, c = rc % HID;
    const float* U = upW + (long long)i * HH;
    const float* D = dnW + (long long)i * HH;
    symW[(long long)(2 * i) * HH + rc]     = 0.5f * (U[r * HID + c] + U[c * HID + r]);
    symW[(long long)(2 * i + 1) * HH + rc] = 0.5f * (D[r * HID + c] + D[c * HID + r]);
  }
}

// Fused per-layer gate: p += au[h]*sigmoid(q); q += ad[h]*sigmoid(p_new)
__global__ void activation_kernel(float* __restrict__ p, float* __restrict__ q,
                                  const float* __restrict__ au,
                                  const float* __restrict__ ad) {
  long long i = (long long)blockIdx.x * blockDim.x + threadIdx.x;
  if (i < (long long)NODES * HID) {
    int h = (int)(i & (HID - 1));
    float pv = p[i], qv = q[i];
    pv += au[h] * (1.0f / (1.0f + __expf(-qv)));
    qv += ad[h] * (1.0f / (1.0f + __expf(-pv)));
    p[i] = pv;
    q[i] = qv;
  }
}

// ------------------------------ WMMA GEMMs ----------------------------------
// C[M,128] (+)= A[M,K] @ B[K,128], row-major, via V_WMMA_F32_16X16X4_F32.
// grid.x = M/16 row tiles; block = (32, 8): each wave owns one 16x16 tile,
// the 8 waves of a block cover the full 128-wide column slab and share the
// A row-tile through L0.
__global__ void gemm_wmma_f32(const float* __restrict__ A,
                              const float* __restrict__ B,
                              float* __restrict__ C,
                              int K, int accum) {
  const int lane = threadIdx.x;        // wave32
  const int l16  = lane & 15;
  const int half = lane >> 4;          // 0: lanes 0-15, 1: lanes 16-31
  const int mt   = blockIdx.x * 16;
  const int nt   = (int)threadIdx.y * 16;   // Nn == HID == 128

  v8f c = {};
  if (accum) {
#pragma unroll
    for (int r = 0; r < 8; ++r)
      c[r] = C[(long long)(mt + half * 8 + r) * HID + nt + l16];
  }

  const float* arow = A + (long long)(mt + l16) * K;
#pragma unroll 8
  for (int k = 0; k < K; k += 4) {
    // A fragment (16x4): lane=M; VGPR pair holds K = {2*half, 2*half+1}
    v2f a;
    a.x = arow[k + 2 * half];
    a.y = arow[k + 2 * half + 1];
    // B fragment (4x16): row striped across lanes within each VGPR
    const int kb = k + 2 * half;
    v2f b;
    b.x = B[(long long)kb * HID + nt + l16];
    b.y = B[(long long)(kb + 1) * HID + nt + l16];
    c = __builtin_amdgcn_wmma_f32_16x16x4_f32(
        /*neg_a=*/false, a, /*neg_b=*/false, b,
        /*c_mod=*/(short)0, c, /*reuse_a=*/false, /*reuse_b=*/false);
  }

#pragma unroll
  for (int r = 0; r < 8; ++r)
    C[(long long)(mt + half * 8 + r) * HID + nt + l16] = c[r];
}

// out[N,40] = Q[N,128] @ Wout[128,40] + bout.  3 column tiles (48), masked.
__global__ void out_gemm_kernel(const float* __restrict__ Q,
                                const float* __restrict__ Wout,
                                const float* __restrict__ bout,
                                float* __restrict__ out) {
  const int lane = threadIdx.x;
  const int l16  = lane & 15;
  const int half = lane >> 4;
  const int mt   = blockIdx.x * 16;
  const int nt   = blockIdx.y * 16;
  const int col  = nt + l16;
  const bool ok  = (col < CLS);

  v8f c = {};
  const float* qrow = Q + (long long)(mt + l16) * HID;
#pragma unroll 8
  for (int k = 0; k < HID; k += 4) {
    v2f a;
    a.x = qrow[k + 2 * half];
    a.y = qrow[k + 2 * half + 1];
    const int kb = k + 2 * half;
    v2f b;
    b.x = ok ? Wout[kb * CLS + col] : 0.0f;
    b.y = ok ? Wout[(kb + 1) * CLS + col] : 0.0f;
    c = __builtin_amdgcn_wmma_f32_16x16x4_f32(
        false, a, false, b, (short)0, c, false, false);
  }

  if (ok) {
    float bias = bout[col];
#pragma unroll
    for (int r = 0; r < 8; ++r)
      out[(long long)(mt + half * 8 + r) * CLS + col] = c[r] + bias;
  }
}

// ------------------------------ launcher ------------------------------------

extern "C" void kernel_launch(void* const* d_in, const int* in_sizes, int n_in,
                              void* d_out, int out_size, void* d_ws, size_t ws_size,
                              hipStream_t stream) {
  const float* x    = (const float*)d_in[0];
  const int*   ei   = (const int*)d_in[1];
  const float* Wp   = (const float*)d_in[2];
  const float* Wq   = (const float*)d_in[3];
  const float* upW  = (const float*)d_in[4];
  const float* dnW  = (const float*)d_in[5];
  const float* aua  = (const float*)d_in[6];
  const float* ada  = (const float*)d_in[7];
  const float* Wout = (const float*)d_in[8];
  const float* bout = (const float*)d_in[9];
  float* out = (float*)d_out;

  const int E = in_sizes[1] / 2;
  const int* src = ei;        // edge_index[0, :]
  const int* dst = ei + E;    // edge_index[1, :]

  const size_t NH = (size_t)NODES * HID;        // 6.4M floats
  float* ws   = (float*)d_ws;
  float* p    = ws;                              // [N,H]
  float* q    = ws + NH;                         // [N,H]
  float* agg  = ws + 2 * NH;                     // [N,H]
  float* symW = ws + 3 * NH;                     // [2*(L+1),H,H]

  const int mtiles  = NODES / 16;                // 3125
  const dim3 gblk(32, 8);                        // 8 waves, one 16x16 tile each
  const int zblocks = (int)((NH + 255) / 256);
  const long long sthreads = (long long)E * HID;
  const int sblocks = (int)((sthreads + 255) / 256);

  // Precompute symmetric weights (tiny).
  symmetrize_kernel<<<((NLAYER + 1) * HID * HID + 255) / 256, 256, 0, stream>>>(
      upW, dnW, symW);

  // p = x @ Wp ; q = x @ Wq   (K = 256)
  gemm_wmma_f32<<<mtiles, gblk, 0, stream>>>(x, Wp, p, FEAT, 0);
  gemm_wmma_f32<<<mtiles, gblk, 0, stream>>>(x, Wq, q, FEAT, 0);

  auto agg_step = [&](const float* vsrc, const float* Wsym, float* dest) {
    zero_kernel<<<zblocks, 256, 0, stream>>>(agg, (int)NH);
    scatter_add_kernel<<<sblocks, 256, 0, stream>>>(vsrc, src, dst, agg, E);
    gemm_wmma_f32<<<mtiles, gblk, 0, stream>>>(agg, Wsym, dest, HID, 1);
  };

  const size_t HH = (size_t)HID * HID;
  for (int i = 0; i < NLAYER; ++i) {
    agg_step(q, symW + (size_t)(2 * i) * HH, p);       // p += agg(q) @ symUp[i]
    agg_step(p, symW + (size_t)(2 * i + 1) * HH, q);   // q += agg(p) @ symDn[i]
    activation_kernel<<<zblocks, 256, 0, stream>>>(p, q, aua + i * HID, ada + i * HID);
  }
  agg_step(q, symW + (size_t)(2 * NLAYER) * HH, p);
  agg_step(p, symW + (size_t)(2 * NLAYER + 1) * HH, q);

  // out = q @ W_out + b_out
  out_gemm_kernel<<<dim3(mtiles, 3), 32, 0, stream>>>(q, Wout, bout, out);
}